// MatcherV3_32976758899086
// MI455X (gfx1250) — compile-verified
//
#include <hip/hip_runtime.h>
#include <math.h>

#define NBOX     1024
#define PI_F     3.141592653f
#define EPS_F    1e-8f
#define IOU_THR  0.1f

typedef float v2f __attribute__((ext_vector_type(2)));
typedef float v8f __attribute__((ext_vector_type(8)));

__device__ __forceinline__ float limit_period_f(float v) {
    return v - floorf(v / (2.0f * PI_F) + 0.5f) * (2.0f * PI_F);
}

// ---------------------------------------------------------------------------
// K0: build shifted boxes, corners, z-extents, volumes
// ---------------------------------------------------------------------------
__global__ void prep_kernel(const float* __restrict__ det_boxes,
                            const float* __restrict__ det_scores,
                            const float* __restrict__ trans,
                            float* __restrict__ boxes,
                            float* __restrict__ scores,
                            float* __restrict__ dirs,
                            float* __restrict__ corners,
                            float* __restrict__ z1a,
                            float* __restrict__ z2a,
                            float* __restrict__ vola) {
    int i = blockIdx.x * blockDim.x + threadIdx.x;
    if (i >= NBOX) return;
    int a = i >> 8;   // agent index (NPER = 256)
    float sx = trans[a * 3 + 0] - trans[0];
    float sy = trans[a * 3 + 1] - trans[1];
    float b0 = det_boxes[i * 7 + 0] + sx;
    float b1 = det_boxes[i * 7 + 1] + sy;
    float b2 = det_boxes[i * 7 + 2];
    float b3 = det_boxes[i * 7 + 3];
    float b4 = det_boxes[i * 7 + 4];
    float b5 = det_boxes[i * 7 + 5];
    float h  = limit_period_f(det_boxes[i * 7 + 6]);
    boxes[i * 7 + 0] = b0; boxes[i * 7 + 1] = b1; boxes[i * 7 + 2] = b2;
    boxes[i * 7 + 3] = b3; boxes[i * 7 + 4] = b4; boxes[i * 7 + 5] = b5;
    boxes[i * 7 + 6] = h;
    scores[i] = det_scores[i];
    dirs[i]   = h;
    float c = cosf(h), s = sinf(h);
    float lx[4] = { 0.5f * b3, -0.5f * b3, -0.5f * b3,  0.5f * b3 };
    float ly[4] = { 0.5f * b4,  0.5f * b4, -0.5f * b4, -0.5f * b4 };
#pragma unroll
    for (int k = 0; k < 4; ++k) {
        corners[i * 8 + k * 2 + 0] = c * lx[k] - s * ly[k] + b0;
        corners[i * 8 + k * 2 + 1] = s * lx[k] + c * ly[k] + b1;
    }
    z1a[i]  = b2 - 0.5f * b5;
    z2a[i]  = b2 + 0.5f * b5;
    vola[i] = b3 * b4 * b5;
}

// ---------------------------------------------------------------------------
// Pair rotated-rect intersection area (register-resident: rank-select instead
// of a data-movement sort, so no dynamically indexed private arrays -> no
// scratch spilling on wave32)
// ---------------------------------------------------------------------------
__device__ __forceinline__ float pair_inter_area(const float* cax, const float* cay,
                                                 const float* cbx, const float* cby) {
    float px[24], py[24];
    bool  vl[24];
#pragma unroll
    for (int k = 0; k < 4; ++k) { px[k] = cax[k];     py[k] = cay[k]; }
#pragma unroll
    for (int k = 0; k < 4; ++k) { px[4 + k] = cbx[k]; py[4 + k] = cby[k]; }

    // corner-inside-polygon tests (both CCW rectangles)
#pragma unroll
    for (int k = 0; k < 4; ++k) {
        bool ina = true, inb = true;
#pragma unroll
        for (int e = 0; e < 4; ++e) {
            int e1 = (e + 1) & 3;
            float ex  = cbx[e1] - cbx[e], ey  = cby[e1] - cby[e];
            float dx  = cax[k]  - cbx[e], dy  = cay[k]  - cby[e];
            ina = ina && (ex * dy - ey * dx >= -EPS_F);
            float ex2 = cax[e1] - cax[e], ey2 = cay[e1] - cay[e];
            float dx2 = cbx[k]  - cax[e], dy2 = cby[k]  - cay[e];
            inb = inb && (ex2 * dy2 - ey2 * dx2 >= -EPS_F);
        }
        vl[k]     = ina;
        vl[4 + k] = inb;
    }

    // 16 edge-pair intersections
#pragma unroll
    for (int ii = 0; ii < 4; ++ii) {
        int i1 = (ii + 1) & 3;
        float a1x = cax[ii], a1y = cay[ii];
        float dax = cax[i1] - a1x, day = cay[i1] - a1y;
#pragma unroll
        for (int jj = 0; jj < 4; ++jj) {
            int j1 = (jj + 1) & 3;
            float b1x = cbx[jj], b1y = cby[jj];
            float dbx = cbx[j1] - b1x, dby = cby[j1] - b1y;
            float den   = dax * dby - day * dbx;
            bool den_ok = fabsf(den) > EPS_F;
            float den_s = den_ok ? den : 1.0f;
            float difx = b1x - a1x, dify = b1y - a1y;
            float t = (difx * dby - dify * dbx) / den_s;
            float u = (difx * day - dify * dax) / den_s;
            bool ok = den_ok && (t >= 0.0f) && (t <= 1.0f) && (u >= 0.0f) && (u <= 1.0f);
            int idx = 8 + ii * 4 + jj;
            px[idx] = a1x + t * dax;
            py[idx] = a1y + t * day;
            vl[idx] = ok;
        }
    }

    // centroid of valid points
    int   cnt = 0;
    float sx = 0.0f, sy = 0.0f;
#pragma unroll
    for (int k = 0; k < 24; ++k) {
        if (vl[k]) { cnt++; sx += px[k]; sy += py[k]; }
    }
    float inv = 1.0f / fmaxf((float)cnt, 1.0f);
    float cx = sx * inv, cy = sy * inv;

    // angles (invalid -> +1e9 sentinel like the reference)
    float ang[24];
#pragma unroll
    for (int k = 0; k < 24; ++k)
        ang[k] = vl[k] ? atan2f(py[k] - cy, px[k] - cx) : 1e9f;

    // stable-sort-equivalent lexicographic rank on (angle, original index)
    int rnk[24];
#pragma unroll
    for (int j = 0; j < 24; ++j) {
        int r = 0;
#pragma unroll
        for (int l = 0; l < 24; ++l)
            r += (ang[l] < ang[j]) || ((ang[l] == ang[j]) && (l < j));
        rnk[j] = r;
    }

    // shoelace over sorted valid ring (positions >= cnt collapse to sp[0]
    // in the reference => zero cross terms; summation order matches sorted
    // positions 0..cnt-1)
    float acc = 0.0f;
#pragma unroll
    for (int p = 0; p < 24; ++p) {
        if (p < cnt) {
            int pn = (p + 1 < cnt) ? (p + 1) : 0;
            float cxp = 0.f, cyp = 0.f, nxx = 0.f, nxy = 0.f;
#pragma unroll
            for (int l = 0; l < 24; ++l) {
                if (rnk[l] == p)  { cxp = px[l]; cyp = py[l]; }
                if (rnk[l] == pn) { nxx = px[l]; nxy = py[l]; }
            }
            acc += (cxp - cx) * (nxy - cy) - (cyp - cy) * (nxx - cx);
        }
    }
    float area = 0.5f * fabsf(acc);
    return (cnt >= 3) ? area : 0.0f;
}

// ---------------------------------------------------------------------------
// K1: IoU3D + adjacency bitmask (1 thread per pair, wave32 ballot -> 1 word)
// ---------------------------------------------------------------------------
__global__ void __launch_bounds__(256)
iou_kernel(const float* __restrict__ corners,
           const float* __restrict__ z1a, const float* __restrict__ z2a,
           const float* __restrict__ vola,
           unsigned int* __restrict__ adjw) {
    int i = blockIdx.y;
    int j = blockIdx.x * blockDim.x + threadIdx.x;

    // CDNA5: global_prefetch_b8 the next row's corners into the caches
    __builtin_prefetch(corners + (size_t)((i + 1) & (NBOX - 1)) * 8, 0, 1);

    float cax[4], cay[4], cbx[4], cby[4];
#pragma unroll
    for (int k = 0; k < 4; ++k) {
        cax[k] = corners[i * 8 + 2 * k];
        cay[k] = corners[i * 8 + 2 * k + 1];
        cbx[k] = corners[j * 8 + 2 * k];
        cby[k] = corners[j * 8 + 2 * k + 1];
    }
    float inter = pair_inter_area(cax, cay, cbx, cby);
    float ih = fmaxf(fminf(z2a[i], z2a[j]) - fmaxf(z1a[i], z1a[j]), 0.0f);
    float iv = inter * ih;
    float iou = iv / fmaxf(vola[i] + vola[j] - iv, 1e-6f);
    bool adj = iou > IOU_THR;
    unsigned long long m = __ballot((int)adj);
    if ((threadIdx.x & 31) == 0)
        adjw[i * 32 + (j >> 5)] = (unsigned int)m;
}

// ---------------------------------------------------------------------------
// K2: sequential greedy clustering — one 1024-thread workgroup, LDS labels
// ---------------------------------------------------------------------------
__global__ void __launch_bounds__(1024)
cluster_kernel(const unsigned int* __restrict__ adjw, int* __restrict__ seg) {
    __shared__ int cl[NBOX];
    __shared__ int nid_s;
    int t = threadIdx.x;
    cl[t] = 0;
    if (t == 0) nid_s = 1;
    __syncthreads();
    for (int i = 0; i < NBOX; ++i) {
        bool newc = (cl[i] == 0);   // uniform read
        int  cur  = nid_s;
        __syncthreads();
        if (newc) {
            unsigned int w = adjw[i * 32 + (t >> 5)];
            if ((w >> (t & 31)) & 1u) cl[t] = cur;
            if (t == 0) nid_s = cur + 1;
        }
        __syncthreads();
    }
    seg[t] = cl[t] - 1;
}

// ---------------------------------------------------------------------------
// K3: per-segment max score -> first argmax index (clipped)
// ---------------------------------------------------------------------------
__global__ void segmax_kernel(const int* __restrict__ seg,
                              const float* __restrict__ scores,
                              int* __restrict__ amax) {
    int s = blockIdx.x * blockDim.x + threadIdx.x;
    if (s >= NBOX) return;
    float smax = -1e30f;
    int am = NBOX;
    for (int j = 0; j < NBOX; ++j) {
        if (seg[j] == s) {
            float sc = scores[j];
            if (sc > smax) { smax = sc; am = j; }
        }
    }
    amax[s] = (am < NBOX) ? am : (NBOX - 1);
}

// ---------------------------------------------------------------------------
// K4: elementwise pass 1 -> VAL columns {scores, scores*m_gt, scores*~m_gt}
// ---------------------------------------------------------------------------
__global__ void pass1_kernel(const int* __restrict__ seg,
                             const float* __restrict__ scores,
                             const float* __restrict__ dirs,
                             const int* __restrict__ amax,
                             int* __restrict__ mgt,
                             float* __restrict__ VAL) {
    int i = blockIdx.x * blockDim.x + threadIdx.x;
    if (i >= NBOX) return;
    int s = seg[i];
    float ref = dirs[amax[s]];
    float dd = fabsf(dirs[i] - ref);
    dd = (dd > PI_F) ? (2.0f * PI_F - dd) : dd;
    int mg = (dd > 0.5f * PI_F) ? 1 : 0;
    mgt[i] = mg;
    float sc = scores[i];
    float v[16];
#pragma unroll
    for (int c = 0; c < 16; ++c) v[c] = 0.0f;
    v[0] = sc;
    v[1] = mg ? sc : 0.0f;
    v[2] = mg ? 0.0f : sc;
#pragma unroll
    for (int c = 0; c < 16; ++c) VAL[i * 16 + c] = v[c];
}

// ---------------------------------------------------------------------------
// K5/K7: segment-sum as one-hot GEMM with V_WMMA_F32_16X16X4_F32.
//   SUM[s, v] = sum_j [seg[j]==s] * VAL[j, v]
// One wave (32 lanes) per 16 segments; exact f32 precision (indicator is 0/1).
// seg (4KB) and VAL (64KB) are staged into LDS with CDNA5 async-to-LDS copies
// (GLOBAL_LOAD_ASYNC_TO_LDS_B128, tracked by ASYNCcnt, fenced with
// s_wait_asynccnt) so the K-loop runs at LDS latency; K is split across two
// independent WMMA accumulators to hide matrix-pipe latency.
// A 16x4 layout: lane M = L&15; V0 = K(half*2+0), V1 = K(half*2+1).
// B 4x16 layout: lane N = L&15; V0 = K(half*2+0), V1 = K(half*2+1).
// C/D: VGPR r holds M = r + half*8, N = L&15.  EXEC all ones (no divergence).
// ---------------------------------------------------------------------------
__global__ void __launch_bounds__(32)
segsum_wmma_kernel(const int* __restrict__ seg,
                   const float* __restrict__ VAL,
                   float* __restrict__ OUT) {
    __shared__ __align__(16) int   sseg[NBOX];        //  4 KB
    __shared__ __align__(16) float sval[NBOX * 16];   // 64 KB (<< 320 KB WGP LDS)

    int lane  = threadIdx.x;
    int n     = lane & 15;
    int half  = lane >> 4;
    int sbase = blockIdx.x * 16;
    int my_s  = sbase + n;     // A-row (M) id == lane&15

    // ---- async stage: one wave issues ASYNCcnt-tracked b128 copies ----
#pragma unroll
    for (int it = 0; it < 8; ++it) {                  // 8*32 lanes*16B = 4 KB
        int e = (it * 32 + lane) * 4;
        unsigned ldsa = (unsigned)(uintptr_t)&sseg[e];
        asm volatile("global_load_async_to_lds_b128 %0, %1, off"
                     :: "v"(ldsa), "v"(seg + e) : "memory");
    }
#pragma unroll 8
    for (int it = 0; it < 128; ++it) {                // 128*32 lanes*16B = 64 KB
        int e = (it * 32 + lane) * 4;
        unsigned ldsa = (unsigned)(uintptr_t)&sval[e];
        asm volatile("global_load_async_to_lds_b128 %0, %1, off"
                     :: "v"(ldsa), "v"(VAL + e) : "memory");
    }
    asm volatile("s_wait_asynccnt 0x0" ::: "memory");
    // single wave per block -> no cross-wave barrier needed

    // ---- K-loop: two independent accumulators over LDS-resident data ----
    v8f c0 = { 0.f, 0.f, 0.f, 0.f, 0.f, 0.f, 0.f, 0.f };
    v8f c1 = { 0.f, 0.f, 0.f, 0.f, 0.f, 0.f, 0.f, 0.f };
#pragma unroll 4
    for (int j0 = 0; j0 < NBOX; j0 += 8) {
        {
            int kb = j0 + half * 2;
            v2f a, b;
            a.x = (sseg[kb]     == my_s) ? 1.0f : 0.0f;
            a.y = (sseg[kb + 1] == my_s) ? 1.0f : 0.0f;
            b.x = sval[(kb + 0) * 16 + n];
            b.y = sval[(kb + 1) * 16 + n];
            c0 = __builtin_amdgcn_wmma_f32_16x16x4_f32(
                     false, a, false, b, (short)0, c0, false, false);
        }
        {
            int kb = j0 + 4 + half * 2;
            v2f a, b;
            a.x = (sseg[kb]     == my_s) ? 1.0f : 0.0f;
            a.y = (sseg[kb + 1] == my_s) ? 1.0f : 0.0f;
            b.x = sval[(kb + 0) * 16 + n];
            b.y = sval[(kb + 1) * 16 + n];
            c1 = __builtin_amdgcn_wmma_f32_16x16x4_f32(
                     false, a, false, b, (short)0, c1, false, false);
        }
    }
    int mbase = sbase + half * 8;
#pragma unroll
    for (int r = 0; r < 8; ++r)
        OUT[(mbase + r) * 16 + n] = c0[r] + c1[r];
}

// ---------------------------------------------------------------------------
// K6: elementwise pass 2 -> direction flip, s_norm, rank/term; build VAL2
// ---------------------------------------------------------------------------
__global__ void pass2_kernel(const int* __restrict__ seg,
                             const float* __restrict__ scores,
                             const float* __restrict__ dirs,
                             const float* __restrict__ boxes,
                             const int* __restrict__ mgt,
                             const float* __restrict__ SUM1,
                             float* __restrict__ VAL) {
    int i = blockIdx.x * blockDim.x + threadIdx.x;
    if (i >= NBOX) return;
    int s = seg[i];
    float ssum = SUM1[s * 16 + 0];
    float sg   = SUM1[s * 16 + 1];
    float sle  = SUM1[s * 16 + 2];
    int  mg   = mgt[i];
    bool flip = (sg <= sle);
    bool add  = flip ? (mg != 0) : (mg == 0);
    float d2    = limit_period_f(dirs[i] + (add ? PI_F : 0.0f));
    float snorm = scores[i] / ((ssum > 0.0f) ? ssum : 1.0f);
    float sci = scores[i];
    int rank = 0;
    for (int j = 0; j < NBOX; ++j) {
        if (seg[j] == s) {
            float sj = scores[j];
            if ((sj > sci) || ((sj == sci) && (j < i))) rank++;
        }
    }
    float term = powf(sci, (float)(rank + 1));
    float v[16];
#pragma unroll
    for (int c = 0; c < 16; ++c) v[c] = 0.0f;
    v[0] = sinf(d2) * snorm;
    v[1] = cosf(d2) * snorm;
#pragma unroll
    for (int c = 0; c < 6; ++c) v[2 + c] = boxes[i * 7 + c] * snorm;
    v[8] = term;
#pragma unroll
    for (int c = 0; c < 16; ++c) VAL[i * 16 + c] = v[c];
}

// ---------------------------------------------------------------------------
// K8: assemble output rows (N x 8)
// ---------------------------------------------------------------------------
__global__ void final_kernel(const float* __restrict__ SUM2,
                             float* __restrict__ out) {
    int s = blockIdx.x * blockDim.x + threadIdx.x;
    if (s >= NBOX) return;
#pragma unroll
    for (int c = 0; c < 6; ++c) out[s * 8 + c] = SUM2[s * 16 + 2 + c];
    out[s * 8 + 6] = atan2f(SUM2[s * 16 + 0], SUM2[s * 16 + 1]);
    out[s * 8 + 7] = fminf(SUM2[s * 16 + 8], 1.0f);
}

// ---------------------------------------------------------------------------
extern "C" void kernel_launch(void* const* d_in, const int* in_sizes, int n_in,
                              void* d_out, int out_size, void* d_ws, size_t ws_size,
                              hipStream_t stream) {
    (void)in_sizes; (void)n_in; (void)out_size; (void)ws_size;
    const float* det_boxes  = (const float*)d_in[0];   // (4,256,7)
    const float* det_scores = (const float*)d_in[1];   // (4,256)
    const float* trans      = (const float*)d_in[2];   // (4,3)
    float* out = (float*)d_out;                        // (1024,8)

    size_t off = 0;
    char* base = (char*)d_ws;
    auto take = [&](size_t bytes) -> void* {
        void* p = base + off;
        off += (bytes + 255) & ~(size_t)255;
        return p;
    };
    float*        boxes   = (float*)take(NBOX * 7 * sizeof(float));
    float*        scores  = (float*)take(NBOX * sizeof(float));
    float*        dirs    = (float*)take(NBOX * sizeof(float));
    float*        corners = (float*)take(NBOX * 8 * sizeof(float));
    float*        z1a     = (float*)take(NBOX * sizeof(float));
    float*        z2a     = (float*)take(NBOX * sizeof(float));
    float*        vola    = (float*)take(NBOX * sizeof(float));
    unsigned int* adjw    = (unsigned int*)take(NBOX * 32 * sizeof(unsigned int));
    int*          seg     = (int*)take(NBOX * sizeof(int));
    int*          amax    = (int*)take(NBOX * sizeof(int));
    int*          mgt     = (int*)take(NBOX * sizeof(int));
    float*        VAL     = (float*)take(NBOX * 16 * sizeof(float));
    float*        SUM1    = (float*)take(NBOX * 16 * sizeof(float));
    float*        SUM2    = (float*)take(NBOX * 16 * sizeof(float));

    prep_kernel<<<4, 256, 0, stream>>>(det_boxes, det_scores, trans,
                                       boxes, scores, dirs, corners, z1a, z2a, vola);
    iou_kernel<<<dim3(4, NBOX), 256, 0, stream>>>(corners, z1a, z2a, vola, adjw);
    cluster_kernel<<<1, 1024, 0, stream>>>(adjw, seg);
    segmax_kernel<<<4, 256, 0, stream>>>(seg, scores, amax);
    pass1_kernel<<<4, 256, 0, stream>>>(seg, scores, dirs, amax, mgt, VAL);
    segsum_wmma_kernel<<<NBOX / 16, 32, 0, stream>>>(seg, VAL, SUM1);
    pass2_kernel<<<4, 256, 0, stream>>>(seg, scores, dirs, boxes, mgt, SUM1, VAL);
    segsum_wmma_kernel<<<NBOX / 16, 32, 0, stream>>>(seg, VAL, SUM2);
    final_kernel<<<4, 256, 0, stream>>>(SUM2, out);
}